// HATS_56925496541933
// MI455X (gfx1250) — compile-verified
//
#include <hip/hip_runtime.h>
#include <hip/hip_bf16.h>

// ---------------- HATS constants (match reference) ----------------
#define HATS_H      180
#define HATS_W      240
#define HATS_K      10
#define HATS_R      3
#define HATS_S      7            // 2R+1
#define HATS_TAU    1000000.0f
#define HATS_DT     100000.0f
#define HATS_GW     (HATS_W / HATS_K)   // 24
#define HATS_GH     (HATS_H / HATS_K)   // 18
#define HATS_NCELLS (HATS_GH * HATS_GW) // 432

typedef __attribute__((ext_vector_type(2))) float v2f;
typedef __attribute__((ext_vector_type(8))) float v8f;

// ---------------- kernel 1: zero output + per-cell counts ----------------
__global__ void hats_init(float* __restrict__ out, float* __restrict__ counts,
                          int n_out, int n_counts) {
    int i = blockIdx.x * blockDim.x + threadIdx.x;
    if (i < n_out)    out[i]    = 0.0f;
    if (i < n_counts) counts[i] = 0.0f;
}

// ---------------- kernel 2: per-(b,cell) valid-event counts ----------------
__global__ void hats_count(const float* __restrict__ ev, const int* __restrict__ lengths,
                           float* __restrict__ counts, int B, int T) {
    int i = blockIdx.x * blockDim.x + threadIdx.x;
    if (i >= B * T) return;
    int b = i / T;
    int e = i - b * T;
    if (e >= lengths[b]) return;
    const float4 E = *reinterpret_cast<const float4*>(ev + ((long)b * T + e) * 4);
    int cell = ((int)E.y / HATS_K) * HATS_GW + (int)E.x / HATS_K;
    atomicAdd(&counts[b * HATS_NCELLS + cell], 1.0f);
}

// ---------------- kernel 3: WMMA pair-tile kernel ----------------
// One wave32 per 16x16 tile of (anchor i, contributor j) pairs.
// exp(-(t_i - t_j)/tau) = exp(-t_i/tau) * exp(+t_j/tau): rank-1 outer product
// via v_wmma_f32_16x16x4_f32 (A column 0 = a_i, B row 0 = b_j).
// Per-event metadata packed into one word: key(cell*2+p)<<18 | y<<9 | x,
// with distinct sentinels for invalid anchors/contributors so a single
// key compare fuses cell + polarity + validity tests.
__launch_bounds__(256)
__global__ void hats_pairs(const float* __restrict__ ev, const int* __restrict__ lengths,
                           float* __restrict__ out, int B, int T) {
    const int lane = threadIdx.x & 31;
    const int wave = threadIdx.x >> 5;
    const int nt   = T >> 4;                 // tiles per axis (128 for T=2048)
    const long tilesPerB = (long)nt * nt;

    long tile = (long)blockIdx.x * 8 + wave; // 8 waves per block
    if (tile >= (long)B * tilesPerB) return;
    int b  = (int)(tile / tilesPerB);
    long r = tile - (long)b * tilesPerB;
    int ti = (int)(r / nt);
    int tj = (int)(r - (long)ti * nt);
    int i0 = ti << 4;                        // anchor rows   (M dimension)
    int j0 = tj << 4;                        // contributors  (N dimension)

    const float* Eb = ev + (long)b * T * 4;
    const int len = lengths[b];
    if (i0 >= len || j0 >= len) return;      // whole tile invalid (uniform)

    // --- uniform temporal band pruning (timestamps sorted per sample) ---
    const float tiMin = Eb[(i0     ) * 4 + 2];
    const float tiMax = Eb[(i0 + 15) * 4 + 2];
    const float tjMin = Eb[(j0     ) * 4 + 2];
    const float tjMax = Eb[(j0 + 15) * 4 + 2];
    if (tiMax < tjMin)           return;     // every pair has dt < 0
    if (tiMin - tjMax > HATS_DT) return;     // every pair has dt > delta_t

    // --- per-wave LDS metadata: slots [0..15]=anchors, [16..31]=contributors ---
    __shared__ __align__(16) uint2 s_md[8][32];   // {packed meta, t bits}

    // every lane loads exactly one event (branchless address select)
    const int eidx = (lane < 16) ? (i0 + lane) : (j0 + (lane - 16));
    const float4 e4 = *reinterpret_cast<const float4*>(Eb + (long)eidx * 4);
    {
        const int xx = (int)e4.x, yy = (int)e4.y, pp = (int)e4.w;
        const unsigned key = (unsigned)((((yy / HATS_K) * HATS_GW + xx / HATS_K) << 1) | pp);
        unsigned pk = (key << 18) | ((unsigned)yy << 9) | (unsigned)xx;
        if (eidx >= len) pk = (lane < 16) ? 0xF0000000u : 0xF8000000u;
        s_md[wave][lane] = make_uint2(pk, __float_as_uint(e4.z));
    }
    __threadfence_block();   // compiler fence; same-wave LDS ops are in-order

    // --- build WMMA operands: A[:,0] = exp(-t_i/tau), B[0,:] = exp(+t_j/tau) ---
    v2f A  = {0.0f, 0.0f};
    v2f Bv = {0.0f, 0.0f};
    if (lane < 16) {
        A.x  = __expf(-e4.z * (1.0f / HATS_TAU));
        const float tc = __uint_as_float(s_md[wave][16 + lane].y);
        Bv.x = __expf(tc * (1.0f / HATS_TAU));
    }

    // --- 16x16 outer product of exp factors: D[M][N] = a_M * b_N ---
    v8f C = {};
    C = __builtin_amdgcn_wmma_f32_16x16x4_f32(
            /*neg_a=*/false, A, /*neg_b=*/false, Bv,
            /*c_mod=*/(short)0, C, /*reuse_a=*/false, /*reuse_b=*/false);

    // C layout: VGPR rr, lane l -> M = rr + 8*(l>=16), N = l&15
    const int N     = lane & 15;
    const int mbase = (lane >> 4) << 3;

    // contributor meta (loop-invariant): one ds_load_b64
    const uint2 mj = s_md[wave][16 + N];
    const unsigned kj = mj.x >> 18;
    const int xj = (int)(mj.x & 0x1FFu);
    const int yj = (int)((mj.x >> 9) & 0x1FFu);
    const float tj_ = __uint_as_float(mj.y);

    // 8 anchor metas = 64 contiguous bytes: 4x ds_load_b128
    const uint4* mb = reinterpret_cast<const uint4*>(&s_md[wave][mbase]);
    uint4 q0 = mb[0], q1 = mb[1], q2 = mb[2], q3 = mb[3];
    unsigned pk_[8] = {q0.x, q0.z, q1.x, q1.z, q2.x, q2.z, q3.x, q3.z};
    unsigned tb_[8] = {q0.y, q0.w, q1.y, q1.w, q2.y, q2.w, q3.y, q3.w};

#pragma unroll
    for (int rr = 0; rr < 8; ++rr) {
        const unsigned mi = pk_[rr];
        const float   ti_ = __uint_as_float(tb_[rr]);
        const float    dt = ti_ - tj_;                       // t_i - t_j
        const unsigned ki = mi >> 18;                        // cell*2+p (or sentinel)
        const unsigned ox = (unsigned)(xj - (int)(mi & 0x1FFu)         + HATS_R);
        const unsigned oy = (unsigned)(yj - (int)((mi >> 9) & 0x1FFu)  + HATS_R);
        const bool m = (ki == kj) && (dt >= 0.0f) && (dt <= HATS_DT) &&
                       (ox <= 2u * HATS_R) && (oy <= 2u * HATS_R);
        if (m) {
            const int idx = (b * (HATS_NCELLS * 2) + (int)ki) * (HATS_S * HATS_S)
                          + (int)oy * HATS_S + (int)ox;
            atomicAdd(&out[idx], C[rr]);
        }
    }
}

// ---------------- kernel 4: normalize by (count + 1e-6) ----------------
__global__ void hats_norm(float* __restrict__ out, const float* __restrict__ counts, int n) {
    int i = blockIdx.x * blockDim.x + threadIdx.x;
    if (i >= n) return;
    out[i] = out[i] / (counts[i / (2 * HATS_S * HATS_S)] + 1e-6f);
}

extern "C" void kernel_launch(void* const* d_in, const int* in_sizes, int n_in,
                              void* d_out, int out_size, void* d_ws, size_t ws_size,
                              hipStream_t stream) {
    const float* ev      = (const float*)d_in[0];  // [B, T, 4]
    const int*   lengths = (const int*)d_in[1];    // [B]
    float*       out     = (float*)d_out;          // [B, 432, 2, 7, 7]
    float*       counts  = (float*)d_ws;           // [B, 432]

    const int B = in_sizes[1];
    const int T = in_sizes[0] / (4 * B);
    const int n_out    = out_size;
    const int n_counts = B * HATS_NCELLS;

    const int thr = 256;
    int n_init = n_out > n_counts ? n_out : n_counts;
    hats_init<<<(n_init + thr - 1) / thr, thr, 0, stream>>>(out, counts, n_out, n_counts);
    hats_count<<<(B * T + thr - 1) / thr, thr, 0, stream>>>(ev, lengths, counts, B, T);

    const int  nt     = T / 16;
    const long waves  = (long)B * nt * nt;
    const int  blocks = (int)((waves + 7) / 8);    // 8 waves (256 thr) per block
    hats_pairs<<<blocks, thr, 0, stream>>>(ev, lengths, out, B, T);

    hats_norm<<<(n_out + thr - 1) / thr, thr, 0, stream>>>(out, counts, n_out);
}